// SimpleRNN_69715909149157
// MI455X (gfx1250) — compile-verified
//
#include <hip/hip_runtime.h>

// ---------------------------------------------------------------------------
// Problem constants (match reference): B=8, T=4096, H=1024, K=128, V=128, OUT=1024
// d_out = outputs (B,T,OUT) f32  ||  final_state (B,K,V) f32
// d_ws  = qkgv (B*T, 512) f32  then  scan-out (B*T, 128) f32   (~80 MB)
// ---------------------------------------------------------------------------
static constexpr int BB   = 8;
static constexpr int TT   = 4096;
static constexpr int HH   = 1024;
static constexpr int KK   = 128;
static constexpr int VV   = 128;
static constexpr int OUTD = 1024;
static constexpr int MM   = BB * TT;      // 32768 rows
static constexpr int NP   = 3 * KK + VV;  // 512 proj cols

typedef __attribute__((ext_vector_type(2))) float v2f;
typedef __attribute__((ext_vector_type(8))) float v8f;

// D = A(16x4 f32) * B(4x16 f32) + C(16x16 f32), full f32 matrix-core path.
// 8-arg form per probe: (neg_a, A, neg_b, B, c_mod, C, reuse_a, reuse_b)
__device__ __forceinline__ v8f wmma_f32x4(v2f a, v2f b, v8f c) {
    return __builtin_amdgcn_wmma_f32_16x16x4_f32(false, a, false, b, (short)0, c,
                                                 false, false);
}

// ---------------------------------------------------------------------------
// Kernel 1: qkgv = hidden(32768,1024) @ W_proj(1024,512) + b_proj,
//           sigmoid on columns [128,384) (k and g groups).
// One wave computes a 16x128 strip (8 N-tiles). grid = (M/16, 512/128).
// A-frag: lane<16 -> rows m0..m0+15 at K=k,k+1 ; lane>=16 -> K=k+2,k+3.
// ---------------------------------------------------------------------------
__global__ void proj_gemm(const float* __restrict__ A, const float* __restrict__ Bm,
                          const float* __restrict__ bias, float* __restrict__ Out) {
    const int lane = threadIdx.x & 31;
    const int lo   = lane & 15;
    const int hi   = lane >> 4;                  // 0 or 1
    const int mt   = blockIdx.x;                 // 16-row tile
    const int ng   = blockIdx.y;                 // 128-col group
    const int n0   = ng * 128;

    const size_t arow = (size_t)(mt * 16 + lo) * HH + (size_t)(hi * 2);

    v8f c[8];
#pragma unroll
    for (int t = 0; t < 8; ++t)
#pragma unroll
        for (int j = 0; j < 8; ++j) c[t][j] = 0.0f;

    for (int k = 0; k < HH; k += 4) {
        v2f a = *(const v2f*)(A + arow + k);  // contiguous pair per lane
        const float* bptr = Bm + (size_t)(k + hi * 2) * NP + n0 + lo;
#pragma unroll
        for (int t = 0; t < 8; ++t) {
            v2f b;
            b.x = bptr[t * 16];        // row k+hi*2
            b.y = bptr[NP + t * 16];   // row k+hi*2+1
            c[t] = wmma_f32x4(a, b, c[t]);
        }
    }

    const bool sig = (ng == 1) || (ng == 2);  // k and g column groups
#pragma unroll
    for (int t = 0; t < 8; ++t) {
        const int col = n0 + t * 16 + lo;
        const float bv = bias[col];
#pragma unroll
        for (int j = 0; j < 8; ++j) {
            const int row = mt * 16 + j + hi * 8;  // C layout: VGPR j -> M=j / j+8
            float v = c[t][j] + bv;
            if (sig) v = 1.0f / (1.0f + __expf(-v));
            Out[(size_t)row * NP + col] = v;
        }
    }
}

// ---------------------------------------------------------------------------
// Kernel 2: gated outer-product scan. grid = (V/8, B), block = 256 (8 waves).
// wave -> one v column; lane -> 4 contiguous k's of state in registers.
// No LDS, no barriers: per-step loads prefetched one step ahead; the only
// serial dependency is the 1-FMA state-update chain.
// ---------------------------------------------------------------------------
__global__ void scan_kernel(const float* __restrict__ qkgv,
                            float* __restrict__ oscan,
                            float* __restrict__ final_state) {
    const int lane = threadIdx.x & 31;
    const int wave = threadIdx.x >> 5;            // 0..7
    const int b    = blockIdx.y;
    const int vcol = blockIdx.x * 8 + wave;       // 0..127

    const float*  base = qkgv + (size_t)b * TT * NP;
    const float4* b4   = (const float4*)base;     // 128 float4 per 512-f row

    float s0 = 0.f, s1 = 0.f, s2 = 0.f, s3 = 0.f;

    // preload t = 0
    float4 q  = b4[lane];
    float4 ks = b4[32 + lane];
    float4 g  = b4[64 + lane];
    float  vv = base[384 + vcol];

    float* orow = oscan + (size_t)b * TT * VV + vcol;

    for (int t = 0; t < TT; ++t) {
        float4 nq = q, nks = ks, ng2 = g;
        float  nvv = vv;
        if (t + 1 < TT) {  // prefetch next step (independent of state chain)
            const float4* nb = b4 + (size_t)(t + 1) * (NP / 4);
            nq  = nb[lane];
            nks = nb[32 + lane];
            ng2 = nb[64 + lane];
            nvv = base[(size_t)(t + 1) * NP + 384 + vcol];
        }
        // state update: cell = cell*g + k*v
        s0 = s0 * g.x + ks.x * vv;
        s1 = s1 * g.y + ks.y * vv;
        s2 = s2 * g.z + ks.z * vv;
        s3 = s3 * g.w + ks.w * vv;
        // readout: o[v] = sum_k q[k]*cell[k,v]  (butterfly over 32 lanes)
        float acc = s0 * q.x + s1 * q.y + s2 * q.z + s3 * q.w;
#pragma unroll
        for (int off = 16; off > 0; off >>= 1) acc += __shfl_xor(acc, off, 32);
        if (lane == 0) orow[(size_t)t * VV] = acc;

        q = nq; ks = nks; g = ng2; vv = nvv;
    }

    // final_state (B,K,V)
    const size_t fs = (size_t)b * KK * VV + (size_t)(lane * 4) * VV + vcol;
    final_state[fs + 0 * VV] = s0;
    final_state[fs + 1 * VV] = s1;
    final_state[fs + 2 * VV] = s2;
    final_state[fs + 3 * VV] = s3;
}

// ---------------------------------------------------------------------------
// Kernel 3: outputs = oscan(32768,128) @ W_out(128,1024) + b_out -> d_out.
// Same 16x128-strip WMMA structure, K=128.
// ---------------------------------------------------------------------------
__global__ void out_gemm(const float* __restrict__ A, const float* __restrict__ Bm,
                         const float* __restrict__ bias, float* __restrict__ Out) {
    const int lane = threadIdx.x & 31;
    const int lo   = lane & 15;
    const int hi   = lane >> 4;
    const int mt   = blockIdx.x;
    const int n0   = blockIdx.y * 128;

    const size_t arow = (size_t)(mt * 16 + lo) * VV + (size_t)(hi * 2);

    v8f c[8];
#pragma unroll
    for (int t = 0; t < 8; ++t)
#pragma unroll
        for (int j = 0; j < 8; ++j) c[t][j] = 0.0f;

    for (int k = 0; k < KK; k += 4) {
        v2f a = *(const v2f*)(A + arow + k);
        const float* bptr = Bm + (size_t)(k + hi * 2) * OUTD + n0 + lo;
#pragma unroll
        for (int t = 0; t < 8; ++t) {
            v2f b;
            b.x = bptr[t * 16];
            b.y = bptr[OUTD + t * 16];
            c[t] = wmma_f32x4(a, b, c[t]);
        }
    }

#pragma unroll
    for (int t = 0; t < 8; ++t) {
        const int col = n0 + t * 16 + lo;
        const float bv = bias[col];
#pragma unroll
        for (int j = 0; j < 8; ++j) {
            const int row = mt * 16 + j + hi * 8;
            Out[(size_t)row * OUTD + col] = c[t][j] + bv;
        }
    }
}

// ---------------------------------------------------------------------------
extern "C" void kernel_launch(void* const* d_in, const int* in_sizes, int n_in,
                              void* d_out, int out_size, void* d_ws, size_t ws_size,
                              hipStream_t stream) {
    const float* hidden = (const float*)d_in[0];  // (B,T,H)
    const float* W_proj = (const float*)d_in[1];  // (H, 3K+V)
    const float* b_proj = (const float*)d_in[2];  // (3K+V,)
    const float* W_out  = (const float*)d_in[3];  // (V, OUT)
    const float* b_out  = (const float*)d_in[4];  // (OUT,)

    float* out         = (float*)d_out;                    // (B,T,OUT) then (B,K,V)
    float* final_state = out + (size_t)MM * OUTD;

    float* qkgv  = (float*)d_ws;                           // (M, 512)  64 MB
    float* oscan = qkgv + (size_t)MM * NP;                 // (M, 128)  16 MB

    // 1) fused projection + bias + sigmoid(k,g)
    proj_gemm<<<dim3(MM / 16, NP / 128), 32, 0, stream>>>(hidden, W_proj, b_proj, qkgv);

    // 2) sequential gated scan (also writes final_state)
    scan_kernel<<<dim3(VV / 8, BB), 256, 0, stream>>>(qkgv, oscan, final_state);

    // 3) output projection + bias
    out_gemm<<<dim3(MM / 16, OUTD / 128), 32, 0, stream>>>(oscan, W_out, b_out, out);
}